// HashPRIFEncoder_62457414418719
// MI455X (gfx1250) — compile-verified
//
#include <hip/hip_runtime.h>
#include <math.h>

// ---------------------------------------------------------------------------
// HashPRIF encoder for MI455X (gfx1250).
//   out[:, 0:3]  = vect = cross(d, cross(origin, d))
//   out[:, 3:35] = hash encode of spherical(d)          (table1)
//   out[:,35:67] = hash encode of spherical(norm(vect)) (table2)
// Strategy: stage each 256KB hash-table level into LDS with the CDNA5
// Tensor Data Mover (one tensor_load_to_lds per block), gather from LDS.
// ---------------------------------------------------------------------------

#define N_LEVELS   16
#define TABLE_SIZE 32768
#define OUT_STRIDE 67
#define HASH_PRIME 2654435761u

typedef __attribute__((ext_vector_type(4))) unsigned int v4u;
typedef __attribute__((ext_vector_type(8))) int          v8i;
typedef __attribute__((ext_vector_type(4))) int          v4i;

// res[l] = floor(16 * (512/16)^(l/15)) = floor(16 * 2^(l/3))
__constant__ float c_res[N_LEVELS] = {
    16.f, 20.f, 25.f, 32.f, 40.f, 50.f, 64.f, 80.f,
    101.f, 128.f, 161.f, 203.f, 256.f, 322.f, 406.f, 512.f};

__device__ __forceinline__ void sph01(float vx, float vy, float vz,
                                      float& u0, float& u1) {
    const float PI_F   = 3.14159265358979323846f;
    const float INV2PI = 0.15915494309189535f;  // 1/(2*pi)
    float z     = fminf(1.f, fmaxf(-1.f, vz));
    float theta = acosf(z);
    float phi   = atan2f(vy, vx);
    u0 = (theta + PI_F) * INV2PI;
    u1 = (phi   + PI_F) * INV2PI;
}

// -------- pass 1: geometry + spherical coords (SoA workspace) --------------
__global__ __launch_bounds__(256) void k_prep(const float* __restrict__ x,
                                              float* __restrict__ out,
                                              float2* __restrict__ wsd,
                                              float2* __restrict__ wsv, int n) {
    int i = blockIdx.x * 256 + threadIdx.x;
    if (i >= n) return;
    const float* xr = x + (size_t)i * 6;
    float ox = xr[0], oy = xr[1], oz = xr[2];
    float dx = xr[3] - ox, dy = xr[4] - oy, dz = xr[5] - oz;
    float inv = 1.f / fmaxf(sqrtf(dx * dx + dy * dy + dz * dz), 1e-12f);
    dx *= inv; dy *= inv; dz *= inv;
    // t = cross(origin, d); vect = cross(d, t)
    float tx = oy * dz - oz * dy;
    float ty = oz * dx - ox * dz;
    float tz = ox * dy - oy * dx;
    float vx = dy * tz - dz * ty;
    float vy = dz * tx - dx * tz;
    float vz = dx * ty - dy * tx;
    float* o = out + (size_t)i * OUT_STRIDE;
    o[0] = vx; o[1] = vy; o[2] = vz;
    float ud0, ud1, uv0, uv1;
    sph01(dx, dy, dz, ud0, ud1);
    float invv = 1.f / fmaxf(sqrtf(vx * vx + vy * vy + vz * vz), 1e-12f);
    sph01(vx * invv, vy * invv, vz * invv, uv0, uv1);
    wsd[i] = make_float2(ud0, ud1);
    wsv[i] = make_float2(uv0, uv1);
}

// -------- pass 2: per-(table,level) encode with LDS-staged table -----------
__global__ __launch_bounds__(512) void k_encode(const float* __restrict__ t1,
                                                const float* __restrict__ t2,
                                                const float2* __restrict__ ws,
                                                float* __restrict__ out,
                                                int n, int ppb) {
    extern __shared__ float2 lds_tbl[];  // 32768 * 8B = 256 KB
    const int tsel  = (int)(blockIdx.y >> 4);
    const int level = (int)(blockIdx.y & 15);
    const float* tb = (tsel ? t2 : t1) + (size_t)level * (TABLE_SIZE * 2);

#if __has_builtin(__builtin_amdgcn_tensor_load_to_lds)
    // Scalarized guard: TDM issues even with EXEC==0, so use a uniform
    // (SGPR) condition to guarantee a scalar branch that only wave 0 takes.
    if (__builtin_amdgcn_readfirstlane((int)(threadIdx.x >> 5)) == 0) {
        unsigned long long ga = (unsigned long long)(uintptr_t)tb;
        unsigned int lds = (unsigned int)(uintptr_t)lds_tbl;  // low 32 = LDS offset
        v4u g0;
        g0[0] = 1u;                                        // count=1, user mode
        g0[1] = lds;                                       // lds_addr
        g0[2] = (unsigned int)ga;                          // global_addr[31:0]
        g0[3] = ((unsigned int)(ga >> 32) & 0x1FFFFFFu)    // global_addr[56:32]
                | (2u << 30);                              // type=2 (image)
        const unsigned int dim0 = 16384, dim1 = 4, tile0 = 16384, tile1 = 4;
        v8i g1;
        g1[0] = (int)(2u << 16);                           // data_size = 4 bytes
        g1[1] = (int)((dim0 & 0xFFFFu) << 16);             // tensor_dim0 lo16
        g1[2] = (int)((dim0 >> 16) | ((dim1 & 0xFFFFu) << 16));
        g1[3] = (int)((dim1 >> 16) | (tile0 << 16));       // tile_dim0
        g1[4] = (int)tile1;                                // tile_dim1, tile_dim2=0
        g1[5] = (int)dim0;                                 // tensor_dim0_stride lo32
        g1[6] = 0;
        g1[7] = 0;
        v4i z4 = {0, 0, 0, 0};
        v8i z8 = {0, 0, 0, 0, 0, 0, 0, 0};
        __builtin_amdgcn_tensor_load_to_lds(g0, g1, z4, z4, z8, 0);
        __builtin_amdgcn_s_wait_tensorcnt(0);
    }
#else
    for (int e = (int)threadIdx.x; e < TABLE_SIZE; e += 512)
        lds_tbl[e] = ((const float2*)tb)[e];
#endif
    __syncthreads();

    const float2* __restrict__ sph = ws + (size_t)tsel * (size_t)n;  // SoA
    const float res = c_res[level];
    int start = (int)blockIdx.x * ppb;
    int end   = start + ppb; if (end > n) end = n;
    for (int i = start + (int)threadIdx.x; i < end; i += 512) {
        float2 u = sph[i];
        float pos0 = u.x * res, pos1 = u.y * res;
        float p0 = floorf(pos0), p1 = floorf(pos1);
        float f0 = pos0 - p0,    f1 = pos1 - p1;
        unsigned c0 = (unsigned)p0, c1 = (unsigned)p1;
        unsigned h0 = c1 * HASH_PRIME, h1 = (c1 + 1u) * HASH_PRIME;
        float2 q00 = lds_tbl[(c0 ^ h0) & (TABLE_SIZE - 1u)];
        float2 q01 = lds_tbl[(c0 ^ h1) & (TABLE_SIZE - 1u)];
        float2 q10 = lds_tbl[((c0 + 1u) ^ h0) & (TABLE_SIZE - 1u)];
        float2 q11 = lds_tbl[((c0 + 1u) ^ h1) & (TABLE_SIZE - 1u)];
        float w00 = (1.f - f0) * (1.f - f1), w01 = (1.f - f0) * f1;
        float w10 = f0 * (1.f - f1),         w11 = f0 * f1;
        float rx = w00 * q00.x + w01 * q01.x + w10 * q10.x + w11 * q11.x;
        float ry = w00 * q00.y + w01 * q01.y + w10 * q10.y + w11 * q11.y;
        float* o = out + (size_t)i * OUT_STRIDE + 3 + tsel * 32 + level * 2;
        o[0] = rx; o[1] = ry;
    }
}

// -------- fallback: fully fused, direct L2 gathers -------------------------
__device__ __forceinline__ float2 lookup_g(const float* __restrict__ tb,
                                           float u0, float u1, float res) {
    float pos0 = u0 * res, pos1 = u1 * res;
    float p0 = floorf(pos0), p1 = floorf(pos1);
    float f0 = pos0 - p0,    f1 = pos1 - p1;
    unsigned c0 = (unsigned)p0, c1 = (unsigned)p1;
    unsigned h0 = c1 * HASH_PRIME, h1 = (c1 + 1u) * HASH_PRIME;
    const float2* t = (const float2*)tb;
    float2 q00 = t[(c0 ^ h0) & (TABLE_SIZE - 1u)];
    float2 q01 = t[(c0 ^ h1) & (TABLE_SIZE - 1u)];
    float2 q10 = t[((c0 + 1u) ^ h0) & (TABLE_SIZE - 1u)];
    float2 q11 = t[((c0 + 1u) ^ h1) & (TABLE_SIZE - 1u)];
    float w00 = (1.f - f0) * (1.f - f1), w01 = (1.f - f0) * f1;
    float w10 = f0 * (1.f - f1),         w11 = f0 * f1;
    return make_float2(w00 * q00.x + w01 * q01.x + w10 * q10.x + w11 * q11.x,
                       w00 * q00.y + w01 * q01.y + w10 * q10.y + w11 * q11.y);
}

__global__ __launch_bounds__(256) void k_fused(const float* __restrict__ x,
                                               const float* __restrict__ t1,
                                               const float* __restrict__ t2,
                                               float* __restrict__ out, int n) {
    int i = blockIdx.x * 256 + threadIdx.x;
    if (i >= n) return;
    const float* xr = x + (size_t)i * 6;
    float ox = xr[0], oy = xr[1], oz = xr[2];
    float dx = xr[3] - ox, dy = xr[4] - oy, dz = xr[5] - oz;
    float inv = 1.f / fmaxf(sqrtf(dx * dx + dy * dy + dz * dz), 1e-12f);
    dx *= inv; dy *= inv; dz *= inv;
    float tx = oy * dz - oz * dy, ty = oz * dx - ox * dz, tz = ox * dy - oy * dx;
    float vx = dy * tz - dz * ty, vy = dz * tx - dx * tz, vz = dx * ty - dy * tx;
    float* o = out + (size_t)i * OUT_STRIDE;
    o[0] = vx; o[1] = vy; o[2] = vz;
    float ud0, ud1, uv0, uv1;
    sph01(dx, dy, dz, ud0, ud1);
    float invv = 1.f / fmaxf(sqrtf(vx * vx + vy * vy + vz * vz), 1e-12f);
    sph01(vx * invv, vy * invv, vz * invv, uv0, uv1);
#pragma unroll
    for (int l = 0; l < N_LEVELS; ++l) {
        float2 fd = lookup_g(t1 + (size_t)l * TABLE_SIZE * 2, ud0, ud1, c_res[l]);
        o[3 + 2 * l]  = fd.x; o[4 + 2 * l]  = fd.y;
        float2 fv = lookup_g(t2 + (size_t)l * TABLE_SIZE * 2, uv0, uv1, c_res[l]);
        o[35 + 2 * l] = fv.x; o[36 + 2 * l] = fv.y;
    }
}

extern "C" void kernel_launch(void* const* d_in, const int* in_sizes, int n_in,
                              void* d_out, int out_size, void* d_ws, size_t ws_size,
                              hipStream_t stream) {
    const float* x  = (const float*)d_in[0];
    const float* t1 = (const float*)d_in[1];
    const float* t2 = (const float*)d_in[2];
    float* out = (float*)d_out;
    int n = in_sizes[0] / 6;
    if (n <= 0) return;

    size_t ws_need = (size_t)n * 4 * sizeof(float);  // two float2 arrays
    if (ws_size >= ws_need) {
        float2* wsd = (float2*)d_ws;       // sph(dir),  n entries
        float2* wsv = wsd + (size_t)n;     // sph(vect), n entries
        k_prep<<<(n + 255) / 256, 256, 0, stream>>>(x, out, wsd, wsv, n);
        const int ppb = 32768;                       // points per encode block
        int chunks = (n + ppb - 1) / ppb;            // 32 chunks x 32 pairs
        dim3 grid((unsigned)chunks, 32);
        k_encode<<<grid, 512, TABLE_SIZE * sizeof(float2), stream>>>(
            t1, t2, (const float2*)d_ws, out, n, ppb);
    } else {
        k_fused<<<(n + 255) / 256, 256, 0, stream>>>(x, t1, t2, out, n);
    }
}